// LIFSpike_89120571392518
// MI455X (gfx1250) — compile-verified
//
#include <hip/hip_runtime.h>
#include <cstdint>
#include <cstddef>

// ---------------------------------------------------------------------------
// LIF spike forward: out[t*N + n] = Heaviside(h_t - 1), h_t = 0.5*v + x_t,
// v <- spike ? 0 : h_t, v0 = 0, T = 4, N = total/4.
//
// Memory-bound streaming kernel (256 MiB traffic, ~11 us floor @ 23.3 TB/s).
// CDNA5 path: async global->LDS b128 DMA (ASYNCcnt) double-buffered pipeline,
// ds_load_b128 consumption, non-temporal b128 stores. wave32 / 8-wave blocks.
// ---------------------------------------------------------------------------

typedef float v4f __attribute__((ext_vector_type(4)));
typedef int   v4i __attribute__((vector_size(16)));  // matches builtin proto

typedef __attribute__((address_space(1))) v4i* global_v4i_ptr;
typedef __attribute__((address_space(3))) v4i* lds_v4i_ptr;

#define T_STEPS 4
#define TPB     256
#define GRID    2048

// ----- gfx1250 async copy: global memory -> LDS, 16 bytes per lane ---------
__device__ __forceinline__ void async_load_b128(void* lds_dst, const void* gsrc) {
#if __has_builtin(__builtin_amdgcn_global_load_async_to_lds_b128)
  __builtin_amdgcn_global_load_async_to_lds_b128(
      (global_v4i_ptr)(void*)gsrc,   // global source (AS1, v4i*)
      (lds_v4i_ptr)lds_dst,          // LDS destination (AS3, v4i*)
      /*offset=*/0, /*cpol=*/0);
#else
  // Fallback: hand-rolled VGLOBAL async op. VDST carries the LDS byte address.
  uint32_t lds_addr =
      (uint32_t)(uintptr_t)(__attribute__((address_space(3))) void*)lds_dst;
  asm volatile("global_load_async_to_lds_b128 %0, %1, off"
               :
               : "v"(lds_addr), "v"((uint64_t)(uintptr_t)gsrc)
               : "memory");
#endif
}

#if __has_builtin(__builtin_amdgcn_s_wait_asynccnt)
#define WAIT_ASYNC(n) __builtin_amdgcn_s_wait_asynccnt(n)
#else
#define WAIT_ASYNC(n) asm volatile("s_wait_asynccnt %0" ::"i"(n) : "memory")
#endif

__global__ __launch_bounds__(TPB) void lif_spike_kernel(
    const float* __restrict__ x, float* __restrict__ out,
    int n4 /* float4 elements per timestep */, int iters) {
  // Double-buffered staging: [buffer][timestep][lane] -> 32 KB of LDS.
  __shared__ v4f buf[2][T_STEPS][TPB];

  const v4f* __restrict__ xv = (const v4f*)x;
  v4f* __restrict__ ov       = (v4f*)out;

  const int tid = threadIdx.x;
  const int nb  = gridDim.x;

  // Issue the 4 timestep slabs of tile `it` into buffer `b` (4 async b128s).
  auto issue = [&](int it, int b) {
    int idx = (it * nb + (int)blockIdx.x) * TPB + tid;
    if (idx >= n4) idx = 0;  // clamp: OOB lanes load a dummy, never stored
#pragma unroll
    for (int t = 0; t < T_STEPS; ++t)
      async_load_b128(&buf[b][t][tid], xv + (size_t)t * n4 + idx);
  };

  issue(0, 0);  // prime the pipeline

  for (int it = 0; it < iters; ++it) {
    if (it + 1 < iters) {
      issue(it + 1, (it + 1) & 1);
      // Async loads complete in order: <=4 outstanding means the current
      // tile's 4 loads (the older group) have landed in LDS.
      WAIT_ASYNC(T_STEPS);
    } else {
      WAIT_ASYNC(0);
    }
    asm volatile("" ::: "memory");  // compiler fence: no hoisting past wait

    const int  idx = (it * nb + (int)blockIdx.x) * TPB + tid;
    const bool ok  = idx < n4;
    const int  b   = it & 1;

    v4f v = {0.0f, 0.0f, 0.0f, 0.0f};  // membrane potential, v0 = 0
#pragma unroll
    for (int t = 0; t < T_STEPS; ++t) {
      v4f xt = buf[b][t][tid];  // ds_load_b128
      v4f s;
#pragma unroll
      for (int c = 0; c < 4; ++c) {
        float h  = fmaf(v[c], 0.5f, xt[c]);  // v - (v-0)/tau + x, tau = 2
        bool  sp = (h >= 1.0f);              // Heaviside(h - vth)
        s[c]     = sp ? 1.0f : 0.0f;
        v[c]     = sp ? 0.0f : h;            // hard reset to 0
      }
      if (ok)  // streamed once, never re-read -> non-temporal b128 store
        __builtin_nontemporal_store(s, ov + (size_t)t * n4 + idx);
    }
  }
}

extern "C" void kernel_launch(void* const* d_in, const int* in_sizes, int n_in,
                              void* d_out, int out_size, void* d_ws,
                              size_t ws_size, hipStream_t stream) {
  const float* x   = (const float*)d_in[0];
  float*       out = (float*)d_out;

  const long long total = in_sizes[0];              // 128*256*32*32 = 33554432
  const int N  = (int)(total / T_STEPS);            // elements per timestep
  const int n4 = N / 4;                             // float4s per timestep

  const long long per_pass = (long long)GRID * TPB; // float4s per iteration
  const int iters = (int)((n4 + per_pass - 1) / per_pass);  // = 4 here

  lif_spike_kernel<<<GRID, TPB, 0, stream>>>(x, out, n4, iters);
}